// CustomDetectionLoss_67929202753868
// MI455X (gfx1250) — compile-verified
//
#include <hip/hip_runtime.h>
#include <hip/hip_bf16.h>

#define NCLS 6

typedef __attribute__((ext_vector_type(2))) float v2f;
typedef __attribute__((ext_vector_type(8))) float v8f;

__device__ __forceinline__ float softplusf(float x) {
    // numerically stable: max(x,0) + log1p(exp(-|x|)); exp arg <= 0 so no overflow
    return fmaxf(x, 0.0f) + log1pf(expf(-fabsf(x)));
}

// Stage 1: one thread per (target, layer) pair. Gather pred[b, 0, gy, gx, :11],
// accumulate (lbox, lobj, lcls), deterministic LDS tree-reduce per block,
// write one partial triple per block to ws (component-major: [3][nb]).
__global__ __launch_bounds__(256)
void yolo_loss_stage1(const float* __restrict__ pred0,
                      const float* __restrict__ pred1,
                      const float* __restrict__ pred2,
                      const float* __restrict__ tgt,
                      float* __restrict__ partial,
                      int n)
{
    __shared__ float sred[3][256];
    const int idx = blockIdx.x * 256 + threadIdx.x;

    float lbox = 0.0f, lobj = 0.0f, lcls = 0.0f;
    if (idx < 3 * n) {
        const int layer = idx / n;
        const int t     = idx - layer * n;
        const float* __restrict__ pred = (layer == 0) ? pred0 : (layer == 1) ? pred1 : pred2;
        const int nx = (layer == 0) ? 160 : (layer == 1) ? 80 : 40;
        const int ny = nx;

        const float* tr = tgt + (size_t)t * 6;
        const int   b  = (int)tr[0];
        const int   c  = (int)tr[1];
        const float tx = tr[2], ty = tr[3], tw = tr[4], th = tr[5];

        int gx = (int)floorf((float)nx * tx); gx = min(max(gx, 0), nx - 1);
        int gy = (int)floorf((float)ny * ty); gy = min(max(gy, 0), ny - 1);

        // pred shape (bs, 3, ny, nx, 11); anchor index fixed at 0
        const size_t off = ((size_t)((b * 3) * ny + gy) * (size_t)nx + (size_t)gx) * 11u;
        const float* __restrict__ p = pred + off;

        lbox = fabsf(p[0] - tx) + fabsf(p[1] - ty) + fabsf(p[2] - tw) + fabsf(p[3] - th);
        lobj = softplusf(-p[4]);
        #pragma unroll
        for (int j = 0; j < NCLS; ++j) {
            const float pc = p[5 + j];
            lcls += softplusf(pc) - ((j == c) ? pc : 0.0f);
        }
    }

    sred[0][threadIdx.x] = lbox;
    sred[1][threadIdx.x] = lobj;
    sred[2][threadIdx.x] = lcls;
    __syncthreads();
    for (int off = 128; off > 0; off >>= 1) {
        if (threadIdx.x < (unsigned)off) {
            sred[0][threadIdx.x] += sred[0][threadIdx.x + off];
            sred[1][threadIdx.x] += sred[1][threadIdx.x + off];
            sred[2][threadIdx.x] += sred[2][threadIdx.x + off];
        }
        __syncthreads();
    }
    if (threadIdx.x == 0) {
        const int nb = gridDim.x;
        partial[0 * nb + blockIdx.x] = sred[0][0];
        partial[1 * nb + blockIdx.x] = sred[1][0];
        partial[2 * nb + blockIdx.x] = sred[2][0];
    }
}

// Stage 2: single wave32. Exact f32 reduction of the per-block partials via
// chained V_WMMA_F32_16X16X4_F32 with B = all-ones (D[m][n] = sum_k A[m][k]).
// Component c occupies A rows 4c..4c+3; its `nb` values stream through the
// K=4 dimension over `passes` chained WMMAs. Branchless masking keeps EXEC
// all-1s at every WMMA (ISA requirement).
__global__ __launch_bounds__(32)
void yolo_loss_stage2(const float* __restrict__ partial,
                      float* __restrict__ out,
                      int nb, int passes, int n)
{
    const int l    = threadIdx.x;   // 0..31
    const int m    = l & 15;        // A row this lane carries (per ISA 16x4 layout)
    const int half = l >> 4;        // 0: K={0,1}, 1: K={2,3}
    const int c    = m >> 2;        // component 0..3 (3 = unused rows 12..15)
    const int rc   = m & 3;         // row index within component
    const int cap  = passes * 4;    // per-row slot capacity
    const int cc   = (c < 3) ? c : 0;
    const float cmask = (c < 3) ? 1.0f : 0.0f;

    v2f ones; ones.x = 1.0f; ones.y = 1.0f;   // B = all-ones (layout-agnostic)
    v8f acc = {0.0f, 0.0f, 0.0f, 0.0f, 0.0f, 0.0f, 0.0f, 0.0f};

    for (int p = 0; p < passes; ++p) {
        const int k0 = half * 2;
        const int i0 = rc * cap + p * 4 + k0;
        const int i1 = i0 + 1;
        // branchless in-range load + mask (v_cndmask, no divergence)
        const float a0 = partial[cc * nb + min(i0, nb - 1)] * ((i0 < nb) ? cmask : 0.0f);
        const float a1 = partial[cc * nb + min(i1, nb - 1)] * ((i1 < nb) ? cmask : 0.0f);
        v2f a; a.x = a0; a.y = a1;
        // 8 args: (neg_a, A, neg_b, B, c_mod, C, reuse_a, reuse_b)
        acc = __builtin_amdgcn_wmma_f32_16x16x4_f32(
            false, a, false, ones, (short)0, acc, false, false);
    }

    // C/D layout: lanes 0-15 hold rows 0-7 in VGPRs 0-7; lanes 16-31 hold rows 8-15.
    const float s03 = acc[0] + acc[1] + acc[2] + acc[3];  // lanes<16: comp0; lanes>=16: comp2
    const float s47 = acc[4] + acc[5] + acc[6] + acc[7];  // lanes<16: comp1; lanes>=16: zeros

    const float scls = __shfl(s03, 16, 32);               // comp2 total from lane 16

    if (l == 0) {
        const float inv_n = 1.0f / (float)((n < 1) ? 1 : n);
        const float lbox = s03  * inv_n * 7.5f;  // BOX_GAIN
        const float lobj = s47  * inv_n * 1.5f;  // DFL_GAIN
        const float lcls = scls * inv_n * 0.5f;  // CLS_GAIN
        out[0] = lbox + lobj + lcls;
        out[1] = lbox;
        out[2] = lobj;
        out[3] = lcls;
    }
}

extern "C" void kernel_launch(void* const* d_in, const int* in_sizes, int n_in,
                              void* d_out, int out_size, void* d_ws, size_t ws_size,
                              hipStream_t stream) {
    const float* pred0 = (const float*)d_in[0];
    const float* pred1 = (const float*)d_in[1];
    const float* pred2 = (const float*)d_in[2];
    const float* tgt   = (const float*)d_in[3];

    const int n     = in_sizes[3] / 6;          // 8192 targets
    const int total = 3 * n;                    // (target, layer) pairs
    const int nb    = (total + 255) / 256;      // 96 blocks

    float* partial = (float*)d_ws;              // [3][nb] floats

    yolo_loss_stage1<<<nb, 256, 0, stream>>>(pred0, pred1, pred2, tgt, partial, n);

    const int passes = (nb + 15) / 16;          // 16 slots/component/WMMA pass
    yolo_loss_stage2<<<1, 32, 0, stream>>>(partial, (float*)d_out, nb, passes, n);
}